// AttentionLSTM_3332894622295
// MI455X (gfx1250) — compile-verified
//
#include <hip/hip_runtime.h>
#include <hip/hip_bf16.h>
#include <math.h>

// Problem dims (compile-time constants from the reference)
#define B_   32
#define T_   256
#define DIN  512
#define DOUT 512
#define DCTX 512
#define AHID 512
#define LCTX 128
#define D4   (4 * DOUT)     // 2048
#define KCAT (DIN + DCTX + DOUT)  // 1536: packed K for fused preact GEMM
#define KSLICES 4

typedef float v2f __attribute__((ext_vector_type(2)));
typedef float v8f __attribute__((ext_vector_type(8)));

// ---------------------------------------------------------------------------
// fp32 WMMA GEMM with pre-transposed B:  C = A[M,K] * BT[N,K]^T
// Each wave computes a 16x64 strip (4 accumulators; one A fragment feeds 4
// WMMAs). 8 waves / 256-thread block. Deterministic split-K over grid.z:
// slice z writes its partial to  Cout + z*sliceStride  (consumer sums).
// All fragment loads are contiguous, 8B-aligned float2 (global_load_b64).
//
// V_WMMA_F32_16X16X4_F32 layouts (ISA 7.12.2, wave32):
//   A 16x4 : lane<16 -> row=lane,    {K0,K1}; lane>=16 -> row=lane-16, {K2,K3}
//   B 4x16 : lane n=lane&15; lo-half lanes hold K0/K1, hi-half K2/K3
//   C 16x16: VGPR v -> row v (lanes 0-15) / v+8 (lanes 16-31), col=lane&15
// ---------------------------------------------------------------------------
__global__ void wmma_gemm_bt(const float* __restrict__ A, int lda,
                             const float* __restrict__ BT, int ldbt,
                             float* __restrict__ Cout, int ldc,
                             long sliceStride,
                             int M, int N, int K, int kSlices)
{
    const int wave = threadIdx.x >> 5;
    const int lane = threadIdx.x & 31;
    const int strip = blockIdx.x * 8 + wave;       // 64-wide N strip index
    const int nStrips = N >> 6;
    if (strip >= nStrips) return;                  // wave-uniform (EXEC stays full)
    const int tileM = blockIdx.y;

    const int r16 = lane & 15;
    const int hi  = lane >> 4;                     // 0: {K0,K1}  1: {K2,K3}

    const int kLen = K / kSlices;
    const int kBeg = blockIdx.z * kLen;
    const int kEnd = kBeg + kLen;

    const float* __restrict__ Arow = A + (long)(tileM * 16 + r16) * lda;
    const int nBase = strip * 64 + r16;
    const float* __restrict__ B0 = BT + (long)(nBase     ) * ldbt;
    const float* __restrict__ B1 = BT + (long)(nBase + 16) * ldbt;
    const float* __restrict__ B2 = BT + (long)(nBase + 32) * ldbt;
    const float* __restrict__ B3 = BT + (long)(nBase + 48) * ldbt;

    v8f acc0 = {}, acc1 = {}, acc2 = {}, acc3 = {};
    for (int k0 = kBeg; k0 < kEnd; k0 += 4) {
        const int ka = k0 + 2 * hi;                // even -> 8B aligned
        const v2f a  = *(const v2f*)(Arow + ka);
        const v2f b0 = *(const v2f*)(B0 + ka);
        const v2f b1 = *(const v2f*)(B1 + ka);
        const v2f b2 = *(const v2f*)(B2 + ka);
        const v2f b3 = *(const v2f*)(B3 + ka);
        acc0 = __builtin_amdgcn_wmma_f32_16x16x4_f32(false, a, false, b0, (short)0, acc0, false, false);
        acc1 = __builtin_amdgcn_wmma_f32_16x16x4_f32(false, a, false, b1, (short)0, acc1, false, false);
        acc2 = __builtin_amdgcn_wmma_f32_16x16x4_f32(false, a, false, b2, (short)0, acc2, false, false);
        acc3 = __builtin_amdgcn_wmma_f32_16x16x4_f32(false, a, false, b3, (short)0, acc3, false, false);
    }

    float* __restrict__ Cs = Cout + (long)blockIdx.z * sliceStride;
    const int colBase = strip * 64 + r16;
#pragma unroll
    for (int v = 0; v < 8; ++v) {
        const long row = tileM * 16 + v + hi * 8;
        Cs[row * ldc + colBase     ] = acc0[v];
        Cs[row * ldc + colBase + 16] = acc1[v];
        Cs[row * ldc + colBase + 32] = acc2[v];
        Cs[row * ldc + colBase + 48] = acc3[v];
    }
}

// ---------------------------------------------------------------------------
// One-time weight transposes (weights stay L2-resident across all 256 steps).
// out[n*K + k] = in[k*N + n]
// ---------------------------------------------------------------------------
__global__ void transpose_kernel(const float* __restrict__ in,
                                 float* __restrict__ out, int K, int N)
{
    const long idx = (long)blockIdx.x * 256 + threadIdx.x;   // over K*N
    const int k = (int)(idx / N);
    const int n = (int)(idx % N);
    out[(long)n * K + k] = in[idx];
}

// WVUT[n*KCAT + k] = concat([W;V;U])^T : k<512 -> W, k<1024 -> V, else U
__global__ void concatT_kernel(const float* __restrict__ W,
                               const float* __restrict__ V,
                               const float* __restrict__ U,
                               float* __restrict__ WVUT)
{
    const long idx = (long)blockIdx.x * 256 + threadIdx.x;   // over 2048*1536
    const int n = (int)(idx / KCAT);
    const int k = (int)(idx % KCAT);
    float v;
    if (k < DIN)            v = W[(long)k * D4 + n];
    else if (k < DIN + DCTX) v = V[(long)(k - DIN) * D4 + n];
    else                     v = U[(long)(k - DIN - DCTX) * D4 + n];
    WVUT[idx] = v;
}

// ---------------------------------------------------------------------------
// Init: h = c = 0; pack A_cat = [x_0 | (wctx, filled later) | h=0]
// grid=(DOUT/256, B)
// ---------------------------------------------------------------------------
__global__ void init_kernel(const float* __restrict__ x,
                            float* __restrict__ h, float* __restrict__ c,
                            float* __restrict__ Acat)
{
    const int b = blockIdx.y;
    const int d = blockIdx.x * 256 + threadIdx.x;
    h[b * DOUT + d] = 0.f;
    c[b * DOUT + d] = 0.f;
    Acat[(long)b * KCAT + d] = x[((long)b * T_) * DIN + d];   // x_0
    Acat[(long)b * KCAT + DIN + DCTX + d] = 0.f;              // h_0
}

// ---------------------------------------------------------------------------
// prj[b,l] = sum_a tanh(h_att[b,a] + att_ctx[b,l,a] + b_att[a]) * w_prj[a]
// h_att arrives as KSLICES split-K partials (summed here -> deterministic).
// One wave32 per (b,l); 8 waves/block; grid = B*L/8.
// ---------------------------------------------------------------------------
__global__ void attn_proj_kernel(const float* __restrict__ h_attP,
                                 const float* __restrict__ att_ctx,
                                 const float* __restrict__ b_att,
                                 const float* __restrict__ w_prj,
                                 float* __restrict__ prj)
{
    const int wave = threadIdx.x >> 5;
    const int lane = threadIdx.x & 31;
    const int idx  = blockIdx.x * 8 + wave;        // (b*L + l)
    const int b    = idx >> 7;
    const float* __restrict__ ha = h_attP + (long)b * AHID;
    const float* __restrict__ ac = att_ctx + (long)idx * AHID;

    float sum = 0.f;
    for (int a = lane; a < AHID; a += 32) {
        float hav = ha[a];
#pragma unroll
        for (int s = 1; s < KSLICES; ++s)
            hav += ha[(long)s * (B_ * AHID) + a];
        sum += tanhf(hav + ac[a] + b_att[a]) * w_prj[a];
    }
#pragma unroll
    for (int off = 16; off > 0; off >>= 1)
        sum += __shfl_xor(sum, off, 32);
    if (lane == 0) prj[idx] = sum;
}

// alpha[b,:] = softmax(prj[b,:]) over L=128. One 128-thread block per b.
__global__ void softmax128_kernel(const float* __restrict__ prj,
                                  float* __restrict__ alpha)
{
    __shared__ float red[LCTX];
    const int b = blockIdx.x, l = threadIdx.x;
    const float v = prj[b * LCTX + l];
    red[l] = v;
    __syncthreads();
    for (int s = 64; s > 0; s >>= 1) {
        if (l < s) red[l] = fmaxf(red[l], red[l + s]);
        __syncthreads();
    }
    const float m = red[0];
    __syncthreads();
    const float e = __expf(v - m);
    red[l] = e;
    __syncthreads();
    for (int s = 64; s > 0; s >>= 1) {
        if (l < s) red[l] += red[l + s];
        __syncthreads();
    }
    alpha[b * LCTX + l] = e / red[0];
}

// wctx[b,c] = sum_l alpha[b,l]*context[b,l,c]; written straight into the
// packed GEMM operand A_cat[:, DIN : DIN+DCTX].  grid=(DCTX/256, B)
__global__ void wctx_kernel(const float* __restrict__ alpha,
                            const float* __restrict__ context,
                            float* __restrict__ Acat)
{
    const int b = blockIdx.y;
    const int cidx = blockIdx.x * 256 + threadIdx.x;
    const float* __restrict__ ctx = context + (long)b * LCTX * DCTX + cidx;
    const float* __restrict__ al  = alpha + b * LCTX;
    float acc = 0.f;
#pragma unroll 4
    for (int l = 0; l < LCTX; ++l)
        acc += al[l] * ctx[(long)l * DCTX];
    Acat[(long)b * KCAT + DIN + cidx] = acc;
}

// ---------------------------------------------------------------------------
// Gates: sum KSLICES preact partials (+bias), update c,h, emit out[:,t,:],
// refresh A_cat's h section, and pack x_{t+1} for the next step.
// grid=(DOUT/256, B)
// ---------------------------------------------------------------------------
__device__ __forceinline__ float sigf(float x) { return 1.f / (1.f + __expf(-x)); }

__global__ void gates_kernel(const float* __restrict__ preactP,
                             const float* __restrict__ bias,
                             const float* __restrict__ x,
                             float* __restrict__ c, float* __restrict__ h,
                             float* __restrict__ Acat,
                             float* __restrict__ out, int t)
{
    const int b = blockIdx.y;
    const int d = blockIdx.x * 256 + threadIdx.x;
    const float* __restrict__ p = preactP + (long)b * D4;
    const long PSL = (long)B_ * D4;                 // slice stride

    float pi = bias[d], pf = bias[DOUT + d], po = bias[2 * DOUT + d],
          pg = bias[3 * DOUT + d];
#pragma unroll
    for (int s = 0; s < KSLICES; ++s) {
        const float* ps = p + (long)s * PSL;
        pi += ps[d];
        pf += ps[DOUT + d];
        po += ps[2 * DOUT + d];
        pg += ps[3 * DOUT + d];
    }
    const float i = sigf(pi), f = sigf(pf), o = sigf(po), g = tanhf(pg);
    const float cn = f * c[b * DOUT + d] + i * g;
    const float hn = o * tanhf(cn);
    c[b * DOUT + d] = cn;
    h[b * DOUT + d] = hn;
    Acat[(long)b * KCAT + DIN + DCTX + d] = hn;     // h section of packed A
    out[((long)b * T_ + t) * DOUT + d] = hn;
    if (t + 1 < T_)                                  // pre-pack x_{t+1}
        Acat[(long)b * KCAT + d] = x[((long)b * T_ + t + 1) * DIN + d];
}

// ---------------------------------------------------------------------------
extern "C" void kernel_launch(void* const* d_in, const int* in_sizes, int n_in,
                              void* d_out, int out_size, void* d_ws, size_t ws_size,
                              hipStream_t stream)
{
    (void)in_sizes; (void)n_in; (void)out_size; (void)ws_size;
    const float* x        = (const float*)d_in[0];   // (B,T,DIN)
    const float* context  = (const float*)d_in[1];   // (B,L,DCTX)
    const float* W        = (const float*)d_in[2];   // (DIN,4D)
    const float* V        = (const float*)d_in[3];   // (DCTX,4D)
    const float* U        = (const float*)d_in[4];   // (DOUT,4D)
    const float* bvec     = (const float*)d_in[5];   // (4D,)
    const float* W_h_att  = (const float*)d_in[6];   // (DOUT,AHID)
    const float* W_ctx    = (const float*)d_in[7];   // (DCTX,AHID)
    const float* b_att    = (const float*)d_in[8];   // (AHID,)
    const float* w_prj    = (const float*)d_in[9];   // (AHID,1)
    float* out = (float*)d_out;                      // (B,T,DOUT)

    // Workspace layout (floats)
    float* ws = (float*)d_ws;
    size_t off = 0;
    float* att_ctx = ws + off;  off += (size_t)B_ * LCTX * AHID;       // 2M
    float* WhT     = ws + off;  off += (size_t)DOUT * AHID;            // 256K
    float* WctxT   = ws + off;  off += (size_t)DCTX * AHID;            // 256K
    float* WVUT    = ws + off;  off += (size_t)D4 * KCAT;              // 3M
    float* h       = ws + off;  off += (size_t)B_ * DOUT;
    float* c       = ws + off;  off += (size_t)B_ * DOUT;
    float* h_attP  = ws + off;  off += (size_t)KSLICES * B_ * AHID;    // 64K
    float* prj     = ws + off;  off += (size_t)B_ * LCTX;
    float* alpha   = ws + off;  off += (size_t)B_ * LCTX;
    float* Acat    = ws + off;  off += (size_t)B_ * KCAT;              // 48K
    float* preactP = ws + off;  off += (size_t)KSLICES * B_ * D4;      // 256K

    // ---- one-time setup: weight transposes, att_ctx precompute, state init
    transpose_kernel<<<dim3((DOUT * AHID) / 256), 256, 0, stream>>>(W_h_att, WhT, DOUT, AHID);
    transpose_kernel<<<dim3((DCTX * AHID) / 256), 256, 0, stream>>>(W_ctx, WctxT, DCTX, AHID);
    concatT_kernel<<<dim3(((long)D4 * KCAT) / 256), 256, 0, stream>>>(W, V, U, WVUT);

    // att_ctx = context @ W_ctx_att  (M=4096, N=512, K=512, single slice)
    wmma_gemm_bt<<<dim3(1, (B_ * LCTX) / 16, 1), 256, 0, stream>>>(
        context, DCTX, WctxT, DCTX, att_ctx, AHID, 0,
        B_ * LCTX, AHID, DCTX, 1);

    init_kernel<<<dim3(DOUT / 256, B_), 256, 0, stream>>>(x, h, c, Acat);

    // ---- recurrence
    for (int t = 0; t < T_; ++t) {
        // 1) h_att partials = h @ W_h_att  (M=32,N=512,K=512, split-K x4)
        wmma_gemm_bt<<<dim3(1, B_ / 16, KSLICES), 256, 0, stream>>>(
            h, DOUT, WhT, DOUT, h_attP, AHID, (long)B_ * AHID,
            B_, AHID, DOUT, KSLICES);
        // 2) prj = tanh(sum(h_attP) + att_ctx + b_att) @ w_att_prj
        attn_proj_kernel<<<dim3((B_ * LCTX) / 8), 256, 0, stream>>>(
            h_attP, att_ctx, b_att, w_prj, prj);
        // 3) alpha = softmax(prj)
        softmax128_kernel<<<dim3(B_), dim3(LCTX), 0, stream>>>(prj, alpha);
        // 4) wctx -> A_cat[:, DIN:DIN+DCTX]
        wctx_kernel<<<dim3(DCTX / 256, B_), 256, 0, stream>>>(alpha, context, Acat);
        // 5) preact partials = [x_t|wctx|h] @ [W;V;U]  (M=32,N=2048,K=1536, x4)
        wmma_gemm_bt<<<dim3((D4 / 64) / 8, B_ / 16, KSLICES), 256, 0, stream>>>(
            Acat, KCAT, WVUT, KCAT, preactP, D4, (long)B_ * D4,
            B_, D4, KCAT, KSLICES);
        // 6) gates -> c,h, out[:,t,:], A_cat h-section, pack x_{t+1}
        gates_kernel<<<dim3(DOUT / 256, B_), 256, 0, stream>>>(
            preactP, bvec, x, c, h, Acat, out, t);
    }
}